// PyG_GCNRegression_35270271435518
// MI455X (gfx1250) — compile-verified
//
#include <hip/hip_runtime.h>

// ---------------------------------------------------------------------------
// PyG GCN regression (3x GCNConv + mean-pool + linear) for MI455X (gfx1250).
//
// Roofline reasoning:
//   * GEMMs: 3 x 1.64 GFLOP fp32  -> microseconds on WMMA pipes; keep exact
//     fp32 via V_WMMA_F32_16X16X4_F32 (no reason to downcast, GEMM is tiny).
//   * Edge aggregation: 3 x 850k edges x 1KB  ~ 2.6 GB of gather+atomic
//     traffic, but h/acc are 25.6 MB each -> fully L2-resident (192 MB L2).
//     This dominates; minimize per-edge work by factoring the symmetric norm:
//        out[d] = dinv[d] * sum_{s in N(d)} (h[s]*dinv[s])
//     -> row-scale in GEMM epilogue, raw scatter-add, fused post-scale.
// ---------------------------------------------------------------------------

typedef __attribute__((ext_vector_type(2))) float v2f;
typedef __attribute__((ext_vector_type(8))) float v8f;

#define HDIM 128

// ---- degree / dinv --------------------------------------------------------
__global__ void k_init_deg(float* deg, int n) {
    int i = blockIdx.x * blockDim.x + threadIdx.x;
    if (i < n) deg[i] = 1.0f;                    // self-loop contributes 1
}

__global__ void k_deg_accum(const long long* __restrict__ dst, int e, float* deg) {
    int i = blockIdx.x * blockDim.x + threadIdx.x;
    if (i < e) atomicAdd(&deg[(int)dst[i]], 1.0f);
}

__global__ void k_dinv(float* deg, int n) {
    int i = blockIdx.x * blockDim.x + threadIdx.x;
    if (i < n) deg[i] = rsqrtf(deg[i]);          // deg >= 1 always
}

__global__ void k_zero(float* p, long long n) {
    long long i = (long long)blockIdx.x * blockDim.x + threadIdx.x;
    if (i < n) p[i] = 0.0f;
}

// ---- GEMM: HS = (X @ W) * dinv[row]   via V_WMMA_F32_16X16X4_F32 ----------
// Block = 256 threads = 8 waves. Block computes a 16x128 strip of HS.
// Wave w owns the 16x16 tile at columns [16w, 16w+16).
// W (64KB) + 16x128 A-strip (8KB) staged in LDS (<< 320KB WGP budget).
__global__ __launch_bounds__(256)
void k_gemm_rowscale(const float* __restrict__ X, const float* __restrict__ W,
                     const float* __restrict__ dinv, float* __restrict__ HS, int n) {
    __shared__ float sW[HDIM * HDIM];   // 64 KB
    __shared__ float sA[16 * HDIM];     //  8 KB

    const int tid  = threadIdx.x;
    const int row0 = blockIdx.x * 16;

    // cooperative float4 load of W: 4096 float4 / 256 threads = 16 each
    {
        const float4* W4  = reinterpret_cast<const float4*>(W);
        float4*       sW4 = reinterpret_cast<float4*>(sW);
        #pragma unroll 4
        for (int t = tid; t < HDIM * HDIM / 4; t += 256) sW4[t] = W4[t];
    }
    // cooperative load of 16x128 A strip (clamp tail rows: keeps EXEC full)
    {
        const float4* X4  = reinterpret_cast<const float4*>(X);
        float4*       sA4 = reinterpret_cast<float4*>(sA);
        #pragma unroll
        for (int t = tid; t < 16 * HDIM / 4; t += 256) {
            int r  = t >> 5;                      // 32 float4 per row
            int c4 = t & 31;
            int gr = row0 + r; if (gr >= n) gr = n - 1;
            sA4[t] = X4[(long long)gr * 32 + c4];
        }
    }
    __syncthreads();

    const int wave = tid >> 5;
    const int lane = tid & 31;
    const int half = lane >> 4;       // 0: K pair {0,1}, 1: K pair {2,3}
    const int m    = lane & 15;
    const int ncol = wave * 16 + m;   // B/C column this lane owns

    v8f c = {};
    #pragma unroll
    for (int k = 0; k < HDIM; k += 4) {
        v2f a, b;
        // A 16x4 layout: v0 = K=(0|2), v1 = K=(1|3), rows on lanes%16
        a.x = sA[m * HDIM + k + 2 * half];
        a.y = sA[m * HDIM + k + 2 * half + 1];
        // B 4x16 layout: rows (K) striped across VGPRs, cols on lanes%16
        b.x = sW[(k + 2 * half)     * HDIM + ncol];
        b.y = sW[(k + 2 * half + 1) * HDIM + ncol];
        c = __builtin_amdgcn_wmma_f32_16x16x4_f32(
                /*neg_a=*/false, a, /*neg_b=*/false, b,
                /*c_mod=*/(short)0, c, /*reuse_a=*/false, /*reuse_b=*/false);
    }

    // C layout: VGPR v -> row (v + 8*half), col = ncol. Row-scale by dinv.
    #pragma unroll
    for (int v = 0; v < 8; ++v) {
        int r = row0 + v + 8 * half;
        if (r < n) HS[(long long)r * HDIM + ncol] = c[v] * dinv[r];
    }
}

// ---- edge scatter: ACC[dst] += HS[src], one wave per edge -----------------
// Lane l moves floats [4l, 4l+4): 512B coalesced b128 gather + 4 f32 atomics.
__global__ __launch_bounds__(256)
void k_aggregate(const float* __restrict__ HS, const long long* __restrict__ src,
                 const long long* __restrict__ dst, int e, int n,
                 float* __restrict__ ACC) {
    long long g   = (long long)blockIdx.x * blockDim.x + threadIdx.x;
    int edge      = (int)(g >> 5);
    int lane      = (int)(g & 31);
    int total     = e + n;                       // real edges + self-loops
    if (edge >= total) return;
    int s, d;
    if (edge < e) { s = (int)src[edge]; d = (int)dst[edge]; }
    else          { s = edge - e;       d = s; }
    float4 v = *(reinterpret_cast<const float4*>(HS + (long long)s * HDIM) + lane);
    float* o = ACC + (long long)d * HDIM + lane * 4;
    atomicAdd(o + 0, v.x);
    atomicAdd(o + 1, v.y);
    atomicAdd(o + 2, v.z);
    atomicAdd(o + 3, v.w);
}

// ---- fused post-scale + bias + ReLU: Y = relu(ACC*dinv[row] + b) ----------
__global__ void k_bias_relu(const float* __restrict__ ACC, const float* __restrict__ dinv,
                            const float* __restrict__ b, float* __restrict__ Y, int n) {
    long long g = (long long)blockIdx.x * blockDim.x + threadIdx.x;
    if (g >= (long long)n * HDIM) return;
    int i = (int)(g >> 7);
    int j = (int)(g & 127);
    float v = ACC[g] * dinv[i] + b[j];
    Y[g] = v > 0.0f ? v : 0.0f;
}

// ---- pooling: pooled[batch[i]] += Y[i], counts[batch[i]] += 1 -------------
__global__ void k_pool(const float* __restrict__ Y, const long long* __restrict__ batch,
                       float* __restrict__ pooled, float* __restrict__ counts, int n) {
    long long g = (long long)blockIdx.x * blockDim.x + threadIdx.x;
    if (g >= (long long)n * HDIM) return;
    int i = (int)(g >> 7);
    int j = (int)(g & 127);
    int gr = (int)batch[i];
    atomicAdd(&pooled[gr * HDIM + j], Y[g]);
    if (j == 0) atomicAdd(&counts[gr], 1.0f);
}

// ---- head: out[g] = dot(pooled[g], Wl)/count + bl  (one wave per graph) ---
__global__ void k_final(const float* __restrict__ pooled, const float* __restrict__ counts,
                        const float* __restrict__ Wl, const float* __restrict__ bl,
                        float* __restrict__ out) {
    int gi   = blockIdx.x;
    int lane = threadIdx.x;
    float p = 0.0f;
    #pragma unroll
    for (int j = lane; j < HDIM; j += 32) p += pooled[gi * HDIM + j] * Wl[j];
    #pragma unroll
    for (int off = 16; off; off >>= 1) p += __shfl_xor(p, off, 32);
    if (lane == 0) out[gi] = p / fmaxf(counts[gi], 1.0f) + bl[0];
}

// ---------------------------------------------------------------------------
extern "C" void kernel_launch(void* const* d_in, const int* in_sizes, int n_in,
                              void* d_out, int out_size, void* d_ws, size_t ws_size,
                              hipStream_t stream) {
    (void)n_in; (void)out_size; (void)ws_size;
    const float*     x     = (const float*)d_in[0];
    const long long* ei    = (const long long*)d_in[1];   // int64 [2,E]
    const long long* batch = (const long long*)d_in[2];   // int64 [N]
    const float* W1 = (const float*)d_in[3];  const float* b1 = (const float*)d_in[4];
    const float* W2 = (const float*)d_in[5];  const float* b2 = (const float*)d_in[6];
    const float* W3 = (const float*)d_in[7];  const float* b3 = (const float*)d_in[8];
    const float* Wl = (const float*)d_in[9];  const float* bl = (const float*)d_in[10];

    const int N = in_sizes[0] / HDIM;     // 50000
    const int E = in_sizes[1] / 2;        // 800000
    const long long* srcp = ei;
    const long long* dstp = ei + E;

    // workspace layout (256B aligned): dinv | bufA | bufB | pooled | counts
    char* ws = (char*)d_ws;
    auto al = [](size_t v) { return (v + 255) & ~(size_t)255; };
    size_t off = 0;
    float* dinv   = (float*)(ws + off); off += al((size_t)N * 4);
    float* bufA   = (float*)(ws + off); off += al((size_t)N * HDIM * 4);
    float* bufB   = (float*)(ws + off); off += al((size_t)N * HDIM * 4);
    float* pooled = (float*)(ws + off);                  // 64*128 floats
    float* counts = pooled + 64 * HDIM;                  // 64 floats, contiguous

    const long long NH = (long long)N * HDIM;

    // degrees -> dinv (in place)
    k_init_deg <<<(N + 255) / 256, 256, 0, stream>>>(dinv, N);
    k_deg_accum<<<(E + 255) / 256, 256, 0, stream>>>(dstp, E, dinv);
    k_dinv     <<<(N + 255) / 256, 256, 0, stream>>>(dinv, N);

    // one GCN layer: GEMM+rowscale -> zero acc -> scatter -> scale+bias+relu
    // result lands in t0 (stream ordering makes the t0/t1 ping-pong safe)
    auto layer = [&](const float* in, const float* W, const float* b,
                     float* t0, float* t1) {
        k_gemm_rowscale<<<(N + 15) / 16, 256, 0, stream>>>(in, W, dinv, t0, N);
        k_zero<<<(unsigned)((NH + 255) / 256), 256, 0, stream>>>(t1, NH);
        long long th = (long long)(E + N) * 32;          // one wave per edge
        k_aggregate<<<(unsigned)((th + 255) / 256), 256, 0, stream>>>(t0, srcp, dstp, E, N, t1);
        k_bias_relu<<<(unsigned)((NH + 255) / 256), 256, 0, stream>>>(t1, dinv, b, t0, N);
    };

    layer(x,    W1, b1, bufA, bufB);   // y -> bufA
    layer(bufA, W2, b2, bufB, bufA);   // y -> bufB
    layer(bufB, W3, b3, bufA, bufB);   // y -> bufA

    // pooling + head
    k_zero <<<(64 * HDIM + 64 + 255) / 256, 256, 0, stream>>>(pooled, 64 * HDIM + 64);
    k_pool <<<(unsigned)((NH + 255) / 256), 256, 0, stream>>>(bufA, batch, pooled, counts, N);
    k_final<<<64, 32, 0, stream>>>(pooled, counts, Wl, bl, (float*)d_out);
}